// ParallelScan_37374805409938
// MI455X (gfx1250) — compile-verified
//
#include <hip/hip_runtime.h>
#include <stdint.h>

// Problem dimensions (fixed by the reference).
constexpr int Bc = 4;
constexpr int Sc = 4096;
constexpr int Dc = 2048;

// Chunked-scan configuration.
constexpr int TPB      = 256;            // 8 waves (wave32) per block
constexpr int VEC      = 4;              // float4 per thread
constexpr int DIMS_PB  = TPB * VEC;      // 1024 dims per block
constexpr int DTILES   = Dc / DIMS_PB;   // 2
constexpr int CHUNKS   = 32;             // chunks along S
constexpr int TLEN     = Sc / CHUNKS;    // 128 timesteps per chunk
constexpr int TILE_T   = 4;              // timesteps staged per async tile
constexpr int NTILES   = TLEN / TILE_T;  // 32 tiles per chunk

// Native clang vector so __builtin_nontemporal_* accepts it, and so loads
// lower to b128.
typedef float v4f __attribute__((ext_vector_type(4)));

__device__ __forceinline__ v4f f4fma(v4f x, v4f y, v4f z) {
  v4f r;
  r.x = fmaf(x.x, y.x, z.x); r.y = fmaf(x.y, y.y, z.y);
  r.z = fmaf(x.z, y.z, z.z); r.w = fmaf(x.w, y.w, z.w);
  return r;
}

// Async copy of 16 bytes per lane from global into LDS (CDNA5 async path,
// tracked by ASYNCcnt). vdst = LDS byte address, vaddr = 64-bit global addr.
__device__ __forceinline__ void async_b128(uint32_t lds_addr, const float* gaddr) {
  asm volatile("global_load_async_to_lds_b128 %0, %1, off"
               :: "v"(lds_addr), "v"(gaddr) : "memory");
}

// ---------------------------------------------------------------------------
// Phase 1: per-chunk local scan (zero init) + chunk aggregates (prodA, hEnd).
// a/b streams staged through LDS with a double-buffered async pipeline.
// Each thread consumes only its own staged column -> no barrier needed,
// only s_wait_asynccnt.
// ---------------------------------------------------------------------------
__global__ __launch_bounds__(TPB)
void scan_phase1(const float* __restrict__ a, const float* __restrict__ b,
                 float* __restrict__ out,
                 float* __restrict__ wsA, float* __restrict__ wsH) {
  __shared__ v4f smA[2][TILE_T][TPB];   // 32 KB
  __shared__ v4f smB[2][TILE_T][TPB];   // 32 KB

  const int tid = threadIdx.x;
  const int dt  = blockIdx.x;   // d-tile
  const int ci  = blockIdx.y;   // chunk
  const int bi  = blockIdx.z;   // batch
  const int d0  = dt * DIMS_PB + tid * VEC;
  const size_t rowBase = ((size_t)bi * Sc + (size_t)ci * TLEN) * Dc + d0;

  auto stage = [&](int buf, int tile) {
    size_t g = rowBase + (size_t)tile * TILE_T * Dc;
#pragma unroll
    for (int r = 0; r < TILE_T; ++r) {
      async_b128((uint32_t)(uintptr_t)&smA[buf][r][tid], a + g);
      async_b128((uint32_t)(uintptr_t)&smB[buf][r][tid], b + g);
      g += Dc;
    }
  };

  v4f h = (v4f){0.f, 0.f, 0.f, 0.f};
  v4f A = (v4f){1.f, 1.f, 1.f, 1.f};

  stage(0, 0);
  for (int tile = 0; tile < NTILES; ++tile) {
    const int buf = tile & 1;
    if (tile + 1 < NTILES) {
      stage(buf ^ 1, tile + 1);                       // 8 new asyncs in flight
      asm volatile("s_wait_asynccnt 8" ::: "memory"); // previous 8 are done
    } else {
      asm volatile("s_wait_asynccnt 0" ::: "memory");
    }
    size_t g = rowBase + (size_t)tile * TILE_T * Dc;
#pragma unroll
    for (int r = 0; r < TILE_T; ++r) {
      v4f av = smA[buf][r][tid];
      v4f bv = smB[buf][r][tid];
      h = f4fma(av, h, bv);       // h = a*h + b
      A = A * av;                 // running product of a
      __builtin_nontemporal_store(h, (v4f*)(out + g));  // streaming store
      g += Dc;
    }
  }

  const size_t wi = ((size_t)bi * CHUNKS + ci) * Dc + d0;
  *(v4f*)(wsA + wi) = A;
  *(v4f*)(wsH + wi) = h;
}

// ---------------------------------------------------------------------------
// Phase 2: serial combine of chunk aggregates per channel; h0 folded in as
// the initial state. wsH is rewritten in place with each chunk's INCOMING
// state. Tiny: B*D/4 = 2048 threads.
// ---------------------------------------------------------------------------
__global__ __launch_bounds__(TPB)
void scan_phase2(const float* __restrict__ h0,
                 const float* __restrict__ wsA, float* __restrict__ wsH) {
  const int idx = blockIdx.x * TPB + threadIdx.x;   // over B*D/VEC
  const int bi  = idx / (Dc / VEC);
  const int d   = (idx % (Dc / VEC)) * VEC;

  v4f state = *(const v4f*)(h0 + (size_t)bi * Dc + d);
#pragma unroll
  for (int c = 0; c < CHUNKS; ++c) {
    const size_t wi = ((size_t)bi * CHUNKS + c) * Dc + d;
    v4f Ac = *(const v4f*)(wsA + wi);
    v4f Hc = *(const v4f*)(wsH + wi);
    *(v4f*)(wsH + wi) = state;           // incoming state for chunk c
    state = f4fma(Ac, state, Hc);        // state = A*state + H
  }
}

// ---------------------------------------------------------------------------
// Phase 3: h_t = h_local_t + (prefix-prod of a up to t) * H_in.
// ---------------------------------------------------------------------------
__global__ __launch_bounds__(TPB)
void scan_phase3(const float* __restrict__ a, float* __restrict__ out,
                 const float* __restrict__ wsH) {
  const int tid = threadIdx.x;
  const int dt  = blockIdx.x;
  const int ci  = blockIdx.y;
  const int bi  = blockIdx.z;
  const int d0  = dt * DIMS_PB + tid * VEC;
  const size_t rowBase = ((size_t)bi * Sc + (size_t)ci * TLEN) * Dc + d0;

  const v4f Hin = *(const v4f*)(wsH + ((size_t)bi * CHUNKS + ci) * Dc + d0);
  v4f P = (v4f){1.f, 1.f, 1.f, 1.f};

  size_t g = rowBase;
#pragma unroll 8
  for (int t = 0; t < TLEN; ++t) {
    v4f av = __builtin_nontemporal_load((const v4f*)(a + g));
    v4f hl = *(const v4f*)(out + g);
    P = P * av;                          // inclusive prefix product
    v4f hv = f4fma(P, Hin, hl);
    __builtin_nontemporal_store(hv, (v4f*)(out + g));
    g += Dc;
  }
}

extern "C" void kernel_launch(void* const* d_in, const int* in_sizes, int n_in,
                              void* d_out, int out_size, void* d_ws, size_t ws_size,
                              hipStream_t stream) {
  const float* a  = (const float*)d_in[0];
  const float* b  = (const float*)d_in[1];
  const float* h0 = (const float*)d_in[2];
  float* out = (float*)d_out;

  float* wsA = (float*)d_ws;                               // B*CHUNKS*D floats
  float* wsH = wsA + (size_t)Bc * CHUNKS * Dc;             // B*CHUNKS*D floats

  dim3 grid1(DTILES, CHUNKS, Bc);                          // 256 blocks
  scan_phase1<<<grid1, TPB, 0, stream>>>(a, b, out, wsA, wsH);

  const int nchan = Bc * Dc / VEC;                         // 2048
  scan_phase2<<<(nchan + TPB - 1) / TPB, TPB, 0, stream>>>(h0, wsA, wsH);

  scan_phase3<<<grid1, TPB, 0, stream>>>(a, out, wsH);
}